// AveragePooling_82437602279963
// MI455X (gfx1250) — compile-verified
//
#include <hip/hip_runtime.h>
#include <stdint.h>

// Sparse average pooling (SCN): out[rules_out[i]] += feats[rules_in[i]] / 8
// Memory-bound scatter-add: ~0.5 GB traffic / 23.3 TB/s => ~21 us floor.
// gfx1250 strategy:
//   * 16 lanes per 256B feature row, float4 (B128) granularity, wave32.
//   * 4-deep ring of global_load_async_to_lds_b128 gathers per wave
//     (ASYNCcnt, in-order completion), th:TH_LOAD_NT so the 384MB one-shot
//     gather stream does not evict the 96MB output resident in the 192MB L2.
//   * Native no-return global_atomic_add_f32 scope:SCOPE_DEV into L2.
//   * Each wave only reads LDS it wrote -> no barriers, waves autonomous.

#define BLOCK_T 256
#define WAVE    32
#define WPB     (BLOCK_T / WAVE)   // 8 waves per block
#define RPT     2                  // rules per wave-tile (32 lanes / 16)
#define F4_PER_ROW 16              // 64 channels / 4
#define CCH     64
#define POOL_SCALE 0.125f
#define DEPTH   3                  // async tiles in flight (ring = DEPTH+1 = 4)
#define RING    4

__global__ __launch_bounds__(BLOCK_T)
void ap_scatter_kernel(const float* __restrict__ feats,
                       const int*   __restrict__ rules_in,
                       const int*   __restrict__ rules_out,
                       float*       __restrict__ out,
                       int n_rules) {
    // [wave][ring buffer][lane] : each lane owns one float4 slot. 16 KB total.
    __shared__ float4 smem[WPB][RING][WAVE];

    const int lane = threadIdx.x & (WAVE - 1);
    const int wv   = threadIdx.x >> 5;
    const int riw  = lane >> 4;        // which of the 2 rules in this wave-tile
    const int sub  = lane & 15;        // float4 index within the 64-ch row

    const int gw     = blockIdx.x * WPB + wv;     // global wave id
    const int nW     = gridDim.x * WPB;           // total waves
    const int ntiles = (n_rules + RPT - 1) / RPT;

    const uint32_t lds_base   = (uint32_t)(uintptr_t)&smem[wv][0][lane];
    const uint32_t lds_bufstr = (uint32_t)sizeof(float4) * WAVE;   // 512 B

    // Issue one async gather tile. Always full-EXEC so ASYNCcnt is uniform;
    // out-of-range rules are clamped to a valid row and never consumed.
    auto issue = [&](int t, int buf) {
        int r  = t * RPT + riw;
        int rc = (r < n_rules) ? r : (n_rules - 1);
        int irow = rules_in[rc];
        uint64_t src = (uint64_t)(uintptr_t)
            ((const float4*)feats + (size_t)irow * F4_PER_ROW + sub);
        uint32_t dst = lds_base + (uint32_t)buf * lds_bufstr;
        asm volatile("global_load_async_to_lds_b128 %0, %1, off th:TH_LOAD_NT"
                     :: "v"(dst), "v"(src) : "memory");
    };

    // Number of tiles this wave owns.
    const int m = (gw < ntiles) ? (ntiles - gw + nW - 1) / nW : 0;

    // Prologue: fill the pipeline.
    const int pre = (m < DEPTH) ? m : DEPTH;
    for (int k = 0; k < pre; ++k) issue(gw + k * nW, k);

    int t = gw;
    for (int j = 0; j < m; ++j, t += nW) {
        const int rem = m - 1 - j;               // tiles left after this one
        if (rem >= DEPTH) {
            issue(t + DEPTH * nW, (j + DEPTH) & (RING - 1));
            // in-order completion: <=DEPTH outstanding => oldest (tile j) done
            asm volatile("s_wait_asynccnt 0x3" ::: "memory");
        } else if (rem == 2) {
            asm volatile("s_wait_asynccnt 0x2" ::: "memory");
        } else if (rem == 1) {
            asm volatile("s_wait_asynccnt 0x1" ::: "memory");
        } else {
            asm volatile("s_wait_asynccnt 0x0" ::: "memory");
        }

        int r = t * RPT + riw;
        if (r < n_rules) {
            int orow = rules_out[r];
            float4 v = smem[wv][j & (RING - 1)][lane];
            float x0 = v.x * POOL_SCALE;
            float x1 = v.y * POOL_SCALE;
            float x2 = v.z * POOL_SCALE;
            float x3 = v.w * POOL_SCALE;
            uint64_t dst = (uint64_t)(uintptr_t)
                (out + (size_t)orow * CCH + sub * 4);
            // Native f32 add atomics, no-return (STOREcnt), RMW at device/L2.
            asm volatile(
                "global_atomic_add_f32 %0, %1, off scope:SCOPE_DEV\n\t"
                "global_atomic_add_f32 %0, %2, off offset:4 scope:SCOPE_DEV\n\t"
                "global_atomic_add_f32 %0, %3, off offset:8 scope:SCOPE_DEV\n\t"
                "global_atomic_add_f32 %0, %4, off offset:12 scope:SCOPE_DEV"
                :: "v"(dst), "v"(x0), "v"(x1), "v"(x2), "v"(x3) : "memory");
        }
    }
}

__global__ __launch_bounds__(BLOCK_T)
void ap_zero_kernel(float* __restrict__ out, int n) {
    int i = blockIdx.x * BLOCK_T + threadIdx.x;
    int n4 = n >> 2;
    if (i < n4) ((float4*)out)[i] = make_float4(0.f, 0.f, 0.f, 0.f);
    if (i == 0) {                      // tail (n % 4 floats), normally empty
        for (int k = n4 * 4; k < n; ++k) out[k] = 0.f;
    }
}

extern "C" void kernel_launch(void* const* d_in, const int* in_sizes, int n_in,
                              void* d_out, int out_size, void* d_ws, size_t ws_size,
                              hipStream_t stream) {
    (void)d_ws; (void)ws_size; (void)n_in;
    const float* feats = (const float*)d_in[0];
    const int*   rin   = (const int*)d_in[1];
    const int*   rout  = (const int*)d_in[2];
    float*       out   = (float*)d_out;
    const int n_rules  = in_sizes[1];

    // 1) zero the output (harness poisons d_out; atomics need a clean base)
    int zthreads = (out_size + 3) / 4;
    int zblocks  = (zthreads + BLOCK_T - 1) / BLOCK_T;
    if (zblocks > 0)
        ap_zero_kernel<<<zblocks, BLOCK_T, 0, stream>>>(out, out_size);

    // 2) gather + scatter-add; 4096 blocks * 8 waves => ~23 tiles per wave,
    //    3 outstanding 512B async gathers per wave for HBM latency hiding.
    int ntiles = (n_rules + RPT - 1) / RPT;
    int blocks_needed = (ntiles + WPB - 1) / WPB;
    int blocks = blocks_needed < 4096 ? blocks_needed : 4096;
    if (blocks > 0)
        ap_scatter_kernel<<<blocks, BLOCK_T, 0, stream>>>(feats, rin, rout,
                                                          out, n_rules);
}